// EMAVectorQuantizer_352187318700
// MI455X (gfx1250) — compile-verified
//
#include <hip/hip_runtime.h>
#include <hip/hip_bf16.h>

// ---------------------------------------------------------------------------
// EMA Vector Quantizer for MI455X (gfx1250, wave32, WMMA + async-to-LDS)
//   z: [65536, 64] f32, codebook E: [8192, 64] f32
//   Dominant cost: 65536x8192x64 f32 GEMM (68.7 GFLOP) -> V_WMMA_F32_16X16X4_F32
//   - 32 rows per wave: two C accumulators share each B load (halves LDS BW/WMMA)
//   - code tiles double-buffered in LDS via GLOBAL_LOAD_ASYNC_TO_LDS (ASYNCcnt)
// ---------------------------------------------------------------------------

#define NCODES   8192
#define CDIM     64
#define NVEC     65536
#define DECAYF   0.99f
#define EPSF     1e-5f
#define COMMITF  0.25f

#define CHUNK    128                 // codes staged per LDS buffer
#define STRIDE   68                  // padded row stride (floats) -> bank-conflict-free b64
#define NSTEPS   (NCODES / CHUNK)    // 64

typedef __attribute__((ext_vector_type(2))) float v2f;
typedef __attribute__((ext_vector_type(8))) float v8f;
typedef __attribute__((ext_vector_type(4))) int   v4i;

#if defined(__has_builtin)
#  if __has_builtin(__builtin_amdgcn_global_load_async_to_lds_b128) && \
      __has_builtin(__builtin_amdgcn_global_load_async_to_lds_b32)  && \
      __has_builtin(__builtin_amdgcn_s_wait_asynccnt)
#    define HAVE_ASYNC_LDS 1
#  endif
#endif
#ifndef HAVE_ASYNC_LDS
#  define HAVE_ASYNC_LDS 0
#endif

#if HAVE_ASYNC_LDS
// Builtin signatures (from hipcc diagnostics): b128 takes (v4i AS1*, v4i AS3*,
// Imm offset, Imm cpol); b32 takes (int AS1*, int AS3*, Imm, Imm).
typedef __attribute__((address_space(1))) v4i* g4p;
typedef __attribute__((address_space(3))) v4i* l4p;
typedef __attribute__((address_space(1))) int* g1p;
typedef __attribute__((address_space(3))) int* l1p;
#endif

// ---- workspace layout (floats) --------------------------------------------
#define WS_E2     0                        // 8192
#define WS_IDX    8192                     // 65536 (int32)
#define WS_CNT    73728                    // 8192
#define WS_DW     81920                    // 524288
#define WS_SSE    606208                   // 1
#define WS_NSUM   606209                   // 1
#define WS_ZERO_N (8192 + 524288 + 2)      // counts + dw + sse + nsum (contiguous)

// ---------------------------------------------------------------------------
__global__ __launch_bounds__(256) void vq_zero_kernel(float* __restrict__ p, int n) {
    int t = blockIdx.x * 256 + threadIdx.x;
    if (t < n) p[t] = 0.0f;
}

__global__ __launch_bounds__(256) void vq_e2_kernel(const float* __restrict__ emb,
                                                    float* __restrict__ e2) {
    int k = blockIdx.x * 256 + threadIdx.x;
    const float* r = emb + (size_t)k * CDIM;
    float s = 0.0f;
#pragma unroll
    for (int d = 0; d < CDIM; ++d) s += r[d] * r[d];
    e2[k] = s;
}

// ---------------------------------------------------------------------------
// Stage one 128-code chunk (rows + ||e||^2) into an LDS buffer.
// Async path: 8 x b128 + 1 x b32 per thread = 9 ASYNCcnt ops per wave
// (threads 128..255 duplicate the e2 loads to the same LDS address with the
//  same value, keeping the per-wave async count uniform).
// ---------------------------------------------------------------------------
__device__ __forceinline__ void stage_chunk(float* __restrict__ smE,
                                            float* __restrict__ smE2,
                                            const float* __restrict__ emb,
                                            const float* __restrict__ e2,
                                            int c0, int t) {
#if HAVE_ASYNC_LDS
#pragma unroll
    for (int j = 0; j < 8; ++j) {
        int q   = t + j * 256;          // float4 index within chunk, 16 per row
        int row = q >> 4;
        int c4  = q & 15;
        __builtin_amdgcn_global_load_async_to_lds_b128(
            (g4p)(emb + (size_t)(c0 + row) * CDIM + c4 * 4),
            (l4p)(smE + row * STRIDE + c4 * 4), 0, 0);
    }
    __builtin_amdgcn_global_load_async_to_lds_b32(
        (g1p)(e2 + c0 + (t & 127)), (l1p)(smE2 + (t & 127)), 0, 0);
#else
#pragma unroll
    for (int j = 0; j < 8; ++j) {
        int q   = t + j * 256;
        int row = q >> 4;
        int c4  = q & 15;
        float4 v = ((const float4*)(emb + (size_t)(c0 + row) * CDIM))[c4];
        *((float4*)(smE + row * STRIDE + c4 * 4)) = v;
    }
    smE2[t & 127] = e2[c0 + (t & 127)];
#endif
}

__device__ __forceinline__ void wait_async_group() {
#if HAVE_ASYNC_LDS
    __builtin_amdgcn_s_wait_asynccnt(9);   // next chunk's 9 ops still in flight
#endif
}

// ---------------------------------------------------------------------------
// Argmin via f32 WMMA. Block = 256 threads = 8 waves; each wave owns 32 z rows
// (two 16x64 A tiles in registers). For each 16-code B tile, one b64 LDS load
// feeds TWO chained v_wmma_f32_16x16x4_f32 accumulators.
// Per-lane running argmax of score = 2*z.e - ||e||^2 (== argmin distance),
// strict '>' + final index tie-break reproduce first-argmin semantics.
// ---------------------------------------------------------------------------
__global__ __launch_bounds__(256) void vq_argmin_kernel(
    const float* __restrict__ z, const float* __restrict__ emb,
    const float* __restrict__ e2, int* __restrict__ idx_out)
{
    __shared__ float smE[2][CHUNK * STRIDE];   // 2 x 34816 B
    __shared__ float smE2[2][CHUNK];           // 2 x 512 B

    const int t    = threadIdx.x;
    const int lane = t & 31;
    const int wave = t >> 5;
    const int hlf  = lane >> 4;    // 0: K pair {0,1}; 1: K pair {2,3} (ISA A layout)
    const int l16  = lane & 15;
    const int m0   = blockIdx.x * 256 + wave * 32;

    // Preload A: 32 rows x 64 K in registers, ISA 16x4 f32 A layout per tile.
    v2f aLo[16], aHi[16];
    {
        const float* zl = z + (size_t)(m0 + l16) * CDIM + 2 * hlf;
        const float* zh = zl + 16 * CDIM;
#pragma unroll
        for (int kc = 0; kc < 16; ++kc) {
            aLo[kc][0] = zl[4 * kc + 0];
            aLo[kc][1] = zl[4 * kc + 1];
            aHi[kc][0] = zh[4 * kc + 0];
            aHi[kc][1] = zh[4 * kc + 1];
        }
    }

    float bestL[8], bestH[8];
    int   idxL[8],  idxH[8];
#pragma unroll
    for (int r = 0; r < 8; ++r) {
        bestL[r] = -3.0e38f; bestH[r] = -3.0e38f;
        idxL[r] = 0; idxH[r] = 0;
    }

    stage_chunk(&smE[0][0], &smE2[0][0], emb, e2, 0, t);

    for (int i = 0; i < NSTEPS; ++i) {
        const int buf = i & 1;
        __syncthreads();                              // prev compute done on buf^1
        stage_chunk(&smE[buf ^ 1][0], &smE2[buf ^ 1][0], emb, e2,
                    ((i + 1) & (NSTEPS - 1)) * CHUNK, t);  // wraparound prefetch
        wait_async_group();                           // my chunk-i ops complete
        __syncthreads();                              // everyone's chunk-i visible
        const int c0 = i * CHUNK;

#pragma unroll
        for (int sub = 0; sub < 8; ++sub) {
            const int cb = sub * 16;
            // B (4x16): lane l16 = column n = code cb+n; VGPR0/1 hold K rows
            // 2*hlf, 2*hlf+1 -> read E[cb+l16] at cols {4kc+2h, 4kc+2h+1}
            const float* brow = &smE[buf][(cb + l16) * STRIDE + 2 * hlf];
            v8f cl = {0.f, 0.f, 0.f, 0.f, 0.f, 0.f, 0.f, 0.f};
            v8f ch = {0.f, 0.f, 0.f, 0.f, 0.f, 0.f, 0.f, 0.f};
#pragma unroll
            for (int kc = 0; kc < 16; ++kc) {
                v2f b;
                b[0] = brow[4 * kc + 0];
                b[1] = brow[4 * kc + 1];
                cl = __builtin_amdgcn_wmma_f32_16x16x4_f32(false, aLo[kc], false, b,
                                                           (short)0, cl, false, false);
                ch = __builtin_amdgcn_wmma_f32_16x16x4_f32(false, aHi[kc], false, b,
                                                           (short)0, ch, false, false);
            }
            const float e2v  = smE2[buf][cb + l16];
            const int   code = c0 + cb + l16;
#pragma unroll
            for (int r = 0; r < 8; ++r) {
                float sl = 2.0f * cl[r] - e2v;
                float sh = 2.0f * ch[r] - e2v;
                if (sl > bestL[r]) { bestL[r] = sl; idxL[r] = code; }
                if (sh > bestH[r]) { bestH[r] = sh; idxH[r] = code; }
            }
        }
    }

    // Cross-lane argmax over the 16 lanes sharing each row
    // (C layout: VGPR r -> row m0 + r + 8*hlf, column = l16).
#pragma unroll
    for (int r = 0; r < 8; ++r) {
        float s = bestL[r]; int bi = idxL[r];
        float u = bestH[r]; int bj = idxH[r];
#pragma unroll
        for (int m = 8; m >= 1; m >>= 1) {
            float s2 = __shfl_xor(s, m, 32);
            int   i2 = __shfl_xor(bi, m, 32);
            float u2 = __shfl_xor(u, m, 32);
            int   j2 = __shfl_xor(bj, m, 32);
            if (s2 > s || (s2 == s && i2 < bi)) { s = s2; bi = i2; }
            if (u2 > u || (u2 == u && j2 < bj)) { u = u2; bj = j2; }
        }
        if (l16 == 0) {
            idx_out[m0 + hlf * 8 + r]      = bi;
            idx_out[m0 + 16 + hlf * 8 + r] = bj;
        }
    }
}

// ---------------------------------------------------------------------------
// Gather z_q, straight-through output, SSE for loss, scatter-add EMA stats.
// ---------------------------------------------------------------------------
__global__ __launch_bounds__(256) void vq_gather_scatter(
    const float* __restrict__ z, const float* __restrict__ emb,
    const int* __restrict__ idx, float* __restrict__ zq_out,
    float* __restrict__ idx_out_f, float* __restrict__ counts,
    float* __restrict__ dw, float* __restrict__ sse)
{
    __shared__ float red[8];
    const int t   = threadIdx.x;
    const int row = blockIdx.x * 4 + (t >> 6);
    const int d   = t & 63;
    const int k   = idx[row];

    const float zv = z[(size_t)row * CDIM + d];
    const float ev = emb[(size_t)k * CDIM + d];
    zq_out[(size_t)row * CDIM + d] = zv + (ev - zv);   // match reference fp arith
    const float diff = ev - zv;

    float p = diff * diff;
#pragma unroll
    for (int m = 16; m >= 1; m >>= 1) p += __shfl_xor(p, m, 32);
    if ((t & 31) == 0) red[t >> 5] = p;
    __syncthreads();
    if (t == 0) {
        float s = 0.0f;
#pragma unroll
        for (int i = 0; i < 8; ++i) s += red[i];
        atomicAdd(sse, s);
    }

    if (d == 0) {
        idx_out_f[row] = (float)k;
        atomicAdd(&counts[k], 1.0f);
    }
    atomicAdd(&dw[(size_t)k * CDIM + d], zv);
}

// ---------------------------------------------------------------------------
__global__ __launch_bounds__(256) void vq_ema_cs(
    const float* __restrict__ ema_cs, const float* __restrict__ counts,
    float* __restrict__ ncs_out, float* __restrict__ nsum)
{
    int k = blockIdx.x * 256 + threadIdx.x;
    float v = ema_cs[k] * DECAYF + counts[k] * (1.0f - DECAYF);
    ncs_out[k] = v;
    float p = v;
#pragma unroll
    for (int m = 16; m >= 1; m >>= 1) p += __shfl_xor(p, m, 32);
    if ((threadIdx.x & 31) == 0) atomicAdd(nsum, p);
}

__global__ __launch_bounds__(256) void vq_finalize(
    const float* __restrict__ ema_emb, const float* __restrict__ dw,
    const float* __restrict__ ncs, const float* __restrict__ nsum,
    const float* __restrict__ sse, float* __restrict__ nee_out,
    float* __restrict__ nemb_out, float* __restrict__ loss_out)
{
    int t = blockIdx.x * 256 + threadIdx.x;   // over K*D = 524288
    int k = t >> 6;
    float nee = ema_emb[t] * DECAYF + dw[t] * (1.0f - DECAYF);
    nee_out[t] = nee;
    float n  = *nsum;
    float cs = (ncs[k] + EPSF) / (n + (float)NCODES * EPSF) * n;
    nemb_out[t] = nee / cs;
    if (t == 0) loss_out[0] = COMMITF * (*sse) / (float)((size_t)NVEC * CDIM);
}

// ---------------------------------------------------------------------------
extern "C" void kernel_launch(void* const* d_in, const int* in_sizes, int n_in,
                              void* d_out, int out_size, void* d_ws, size_t ws_size,
                              hipStream_t stream) {
    const float* z       = (const float*)d_in[0];   // [65536,64]
    const float* emb     = (const float*)d_in[1];   // [8192,64]
    const float* ema_cs  = (const float*)d_in[2];   // [8192]
    const float* ema_emb = (const float*)d_in[3];   // [8192,64]

    float* out    = (float*)d_out;
    float* o_zq   = out;                               // 4194304
    float* o_idx  = out + 4194304;                     // 65536
    float* o_loss = out + 4194304 + 65536;             // 1
    float* o_nemb = o_loss + 1;                        // 524288
    float* o_ncs  = o_nemb + 524288;                   // 8192
    float* o_nee  = o_ncs + 8192;                      // 524288

    float* ws     = (float*)d_ws;                      // needs ~2.4 MB
    float* w_e2   = ws + WS_E2;
    int*   w_idx  = (int*)(ws + WS_IDX);
    float* w_cnt  = ws + WS_CNT;
    float* w_dw   = ws + WS_DW;
    float* w_sse  = ws + WS_SSE;
    float* w_nsum = ws + WS_NSUM;

    vq_zero_kernel<<<(WS_ZERO_N + 255) / 256, 256, 0, stream>>>(w_cnt, WS_ZERO_N);
    vq_e2_kernel<<<NCODES / 256, 256, 0, stream>>>(emb, w_e2);
    vq_argmin_kernel<<<NVEC / 256, 256, 0, stream>>>(z, emb, w_e2, w_idx);
    vq_gather_scatter<<<NVEC / 4, 256, 0, stream>>>(z, emb, w_idx, o_zq, o_idx,
                                                    w_cnt, w_dw, w_sse);
    vq_ema_cs<<<NCODES / 256, 256, 0, stream>>>(ema_cs, w_cnt, o_ncs, w_nsum);
    vq_finalize<<<(NCODES * CDIM) / 256, 256, 0, stream>>>(ema_emb, w_dw, o_ncs,
                                                           w_nsum, w_sse, o_nee,
                                                           o_nemb, o_loss);
}